// STGAT_like_Classifier_15307263443463
// MI455X (gfx1250) — compile-verified
//
#include <hip/hip_runtime.h>
#include <hip/hip_bf16.h>
#include <math.h>

#define T_STEPS 20
#define HID     32
#define BATCH   2048
#define NAG     16
#define BN      (BATCH * NAG)     /* 32768 */
#define MG      (BATCH * T_STEPS) /* 40960 */

typedef float v2f __attribute__((ext_vector_type(2)));
typedef float v8f __attribute__((ext_vector_type(8)));

__device__ __forceinline__ float sigm(float x) { return 1.f / (1.f + __expf(-x)); }

__device__ __forceinline__ void lds_fence() {
    asm volatile("s_wait_dscnt 0" ::: "memory");
}

// ---- fp32 WMMA 16x16 tile: acc += A(16xK) @ B(Kx16), row-major LDS operands ----
// A-frag (16x4 f32): lane L holds row M=L&15, K = 4*kt + 2*(L>>4) + v  -> b64 load
// B-frag (4x16 f32): lane L holds col N=L&15, K = 4*kt + 2*(L>>4) + v  -> 2x b32
__device__ __forceinline__ v8f wmma_acc16(v8f acc, const float* __restrict__ A, int lda,
                                          const float* __restrict__ B, int ldb, int K) {
    const int lane = threadIdx.x & 31;
    const int l15  = lane & 15;
    const int hi   = lane >> 4;
    for (int k = 0; k < K; k += 4) {
        v2f a = *(const v2f*)(A + l15 * lda + k + 2 * hi);
        v2f b;
        b.x = B[(k + 2 * hi    ) * ldb + l15];
        b.y = B[(k + 2 * hi + 1) * ldb + l15];
        acc = __builtin_amdgcn_wmma_f32_16x16x4_f32(false, a, false, b, (short)0, acc,
                                                    false, false);
    }
    return acc;
}

// C-layout (16x16 f32): VGPR r -> row r + 8*(L>>4), col = L&15
__device__ __forceinline__ void store_tile16(float* C, int ldc, v8f acc) {
    const int lane = threadIdx.x & 31;
    const int l15 = lane & 15, hi = lane >> 4;
#pragma unroll
    for (int r = 0; r < 8; ++r)
        C[(r + 8 * hi) * ldc + l15] = acc[r];
}

// C(16xN) = A(16xK) @ B(KxN); K % 4 == 0, N % 16 == 0
__device__ __forceinline__ void matmul16(const float* A, int lda, const float* B, int ldb,
                                         float* C, int ldc, int K, int N) {
    for (int nt = 0; nt < N; nt += 16) {
        v8f acc = {0.f, 0.f, 0.f, 0.f, 0.f, 0.f, 0.f, 0.f};
        acc = wmma_acc16(acc, A, lda, B + nt, ldb, K);
        store_tile16(C + nt, ldc, acc);
    }
}

// ============================ LSTM (temporal & graph) ============================
// gates(16x128) = [h(32) | x(DIN) | 1 | 0pad](16xKP) @ W'(KPx128), W' block-shared.
template <int DIN>
__global__ void __launch_bounds__(64) lstm_kernel(
    const float* __restrict__ xin, int xRowStride,
    const float* __restrict__ h0, const float* __restrict__ c0,
    const float* __restrict__ Wih, const float* __restrict__ Whh,
    const float* __restrict__ bih, const float* __restrict__ bhh,
    float* __restrict__ hs) {
    constexpr int KP = (32 + DIN + 2 + 3) & ~3; // 36 (D=2) or 68 (D=32)
    __shared__ __align__(16) float Wp[KP * 128];
    __shared__ __align__(16) float hx[2][16 * KP];
    __shared__ __align__(16) float cb[2][512];
    __shared__ __align__(16) float gt[2][2048];

    const int tid = threadIdx.x;
    const int w = tid >> 5, lane = tid & 31;
    const int base = (blockIdx.x * 2 + w) * 16;

    // Build W' = [Whh ; Wih ; bias ; 0] (KP x 128)
    for (int idx = tid; idx < KP * 128; idx += 64) {
        int k = idx >> 7, j = idx & 127;
        float v;
        if (k < 32)             v = Whh[k * 128 + j];
        else if (k < 32 + DIN)  v = Wih[(k - 32) * 128 + j];
        else if (k == 32 + DIN) v = bih[j] + bhh[j];
        else                    v = 0.f;
        Wp[idx] = v;
    }
    // init hx rows: [h0 | x-slots | 1 | 0], and c
    for (int idx = lane; idx < 16 * KP; idx += 32) {
        int m = idx / KP, k = idx % KP;
        float v;
        if (k < 32)             v = h0[(size_t)(base + m) * 32 + k];
        else if (k == 32 + DIN) v = 1.f;
        else                    v = 0.f;
        hx[w][idx] = v;
    }
    for (int idx = lane; idx < 512; idx += 32)
        cb[w][idx] = c0[(size_t)(base + (idx >> 5)) * 32 + (idx & 31)];
    __syncthreads();

    for (int t = 0; t < T_STEPS; ++t) {
        // stage x_t into hx[:, 32:32+DIN]
        for (int idx = lane; idx < 16 * DIN; idx += 32) {
            int m = idx / DIN, k = idx % DIN;
            hx[w][m * KP + 32 + k] =
                xin[(size_t)(base + m) * xRowStride + t * DIN + k];
        }
        lds_fence();
        matmul16(&hx[w][0], KP, Wp, 128, &gt[w][0], 128, KP, 128);
        lds_fence();
        // LSTM cell (gate order i,f,g,o)
        for (int idx = lane; idx < 512; idx += 32) {
            int m = idx >> 5, j = idx & 31;
            const float* g = &gt[w][m * 128];
            float ig = sigm(g[j]);
            float fg = sigm(g[j + 32]);
            float gg = tanhf(g[j + 64]);
            float og = sigm(g[j + 96]);
            float c = fg * cb[w][idx] + ig * gg;
            float h = og * tanhf(c);
            cb[w][idx] = c;
            hx[w][m * KP + j] = h;
            hs[((size_t)t * BN + base + m) * HID + j] = h;
        }
        lds_fence();
    }
}

// ================================ GAT encoder ================================
__global__ void __launch_bounds__(128) gat_kernel(
    const float* __restrict__ hs1,
    const float* __restrict__ w0, const float* __restrict__ as0,
    const float* __restrict__ ad0, const float* __restrict__ b0,
    const float* __restrict__ w1, const float* __restrict__ as1,
    const float* __restrict__ ad1, const float* __restrict__ b1,
    float* __restrict__ graph) {
    __shared__ __align__(16) float W0s[2048]; // (4,32,16)
    __shared__ __align__(16) float W1s[2048]; // (64,32)
    __shared__ float A0s[64], D0s[64], A1s[32], D1s[32], B0s[16], B1s[32];
    __shared__ __align__(16) float Y[4][1024];
    __shared__ __align__(16) float S[4][1024];
    __shared__ __align__(16) float ATb[4][256];
    __shared__ float V[4][32];

    const int tid = threadIdx.x, w = tid >> 5, lane = tid & 31;
    for (int i = tid; i < 2048; i += 128) { W0s[i] = w0[i]; W1s[i] = w1[i]; }
    if (tid < 64) { A0s[tid] = as0[tid]; D0s[tid] = ad0[tid]; }
    if (tid < 32) { A1s[tid] = as1[tid]; D1s[tid] = ad1[tid]; B1s[tid] = b1[tid]; }
    if (tid < 16) B0s[tid] = b0[tid];
    __syncthreads();

    const int mg = blockIdx.x * 4 + w;       // graph id in [0, MG)
    const int bat = mg / T_STEPS, t = mg % T_STEPS;

    // y = g_in (16 x 32)
    for (int idx = lane; idx < 512; idx += 32)
        Y[w][idx] = hs1[((size_t)t * BN + (size_t)bat * 16 + (idx >> 5)) * 32 + (idx & 31)];
    lds_fence();
    // instance norm over node dim; feature = lane
    {
        float s = 0.f, s2 = 0.f;
        for (int na = 0; na < 16; ++na) { float v = Y[w][na * 32 + lane]; s += v; s2 += v * v; }
        float mean = s * (1.f / 16.f);
        float rs = rsqrtf(s2 * (1.f / 16.f) - mean * mean + 1e-5f);
        for (int na = 0; na < 16; ++na)
            Y[w][na * 32 + lane] = (Y[w][na * 32 + lane] - mean) * rs;
    }
    lds_fence();
    // hp_h = y @ w0[h]  for all heads
    for (int h = 0; h < 4; ++h)
        matmul16(&Y[w][0], 32, &W0s[h * 512], 16, &S[w][h * 256], 16, 32, 16);
    lds_fence();
    // per-head attention, out -> y2 (Y, stride 64, cols h*16..)
    for (int h = 0; h < 4; ++h) {
        {
            int which = lane >> 4, na = lane & 15;
            const float* av = which ? &D0s[h * 16] : &A0s[h * 16];
            float s = 0.f;
            for (int o = 0; o < 16; ++o) s += S[w][h * 256 + na * 16 + o] * av[o];
            V[w][which * 16 + na] = s;
        }
        lds_fence();
        if (lane < 16) {
            float e[16], mx = -3.4e38f;
            float si = V[w][lane];
            for (int j = 0; j < 16; ++j) {
                float v = si + V[w][16 + j];
                v = v > 0.f ? v : 0.2f * v; // leaky relu 0.2
                e[j] = v; mx = fmaxf(mx, v);
            }
            float sum = 0.f;
            for (int j = 0; j < 16; ++j) { e[j] = __expf(e[j] - mx); sum += e[j]; }
            float inv = 1.f / sum;
            for (int j = 0; j < 16; ++j) ATb[w][lane * 16 + j] = e[j] * inv;
        }
        lds_fence();
        matmul16(&ATb[w][0], 16, &S[w][h * 256], 16, &Y[w][h * 16], 64, 16, 16);
        lds_fence();
    }
    // elu(out + b0) -> y2
    for (int idx = lane; idx < 1024; idx += 32) {
        float v = Y[w][idx] + B0s[idx & 15];
        Y[w][idx] = v > 0.f ? v : expm1f(v);
    }
    lds_fence();
    // instance norm over nodes, 64 features
    for (int f = lane; f < 64; f += 32) {
        float s = 0.f, s2 = 0.f;
        for (int na = 0; na < 16; ++na) { float v = Y[w][na * 64 + f]; s += v; s2 += v * v; }
        float mean = s * (1.f / 16.f);
        float rs = rsqrtf(s2 * (1.f / 16.f) - mean * mean + 1e-5f);
        for (int na = 0; na < 16; ++na)
            Y[w][na * 64 + f] = (Y[w][na * 64 + f] - mean) * rs;
    }
    lds_fence();
    // hp1 = y2 @ w1  (16x64 @ 64x32)
    matmul16(&Y[w][0], 64, &W1s[0], 32, &S[w][0], 32, 64, 32);
    lds_fence();
    {
        int which = lane >> 4, na = lane & 15;
        const float* av = which ? D1s : A1s;
        float s = 0.f;
        for (int o = 0; o < 32; ++o) s += S[w][na * 32 + o] * av[o];
        V[w][which * 16 + na] = s;
    }
    lds_fence();
    if (lane < 16) {
        float e[16], mx = -3.4e38f;
        float si = V[w][lane];
        for (int j = 0; j < 16; ++j) {
            float v = si + V[w][16 + j];
            v = v > 0.f ? v : 0.2f * v;
            e[j] = v; mx = fmaxf(mx, v);
        }
        float sum = 0.f;
        for (int j = 0; j < 16; ++j) { e[j] = __expf(e[j] - mx); sum += e[j]; }
        float inv = 1.f / sum;
        for (int j = 0; j < 16; ++j) ATb[w][lane * 16 + j] = e[j] * inv;
    }
    lds_fence();
    matmul16(&ATb[w][0], 16, &S[w][0], 32, &Y[w][0], 32, 16, 32);
    lds_fence();
    // graph[(b*16+na)][t][:] = out + b1
    for (int idx = lane; idx < 512; idx += 32) {
        int na = idx >> 5, j = idx & 31;
        graph[((size_t)bat * 16 + na) * (T_STEPS * HID) + (size_t)t * 32 + j] =
            Y[w][idx] + B1s[j];
    }
}

// ================================ Head (gmlp + classifier) ================================
__global__ void __launch_bounds__(128) head_kernel(
    const float* __restrict__ hs1, const float* __restrict__ hs2,
    const float* __restrict__ x_pair,
    const float* __restrict__ gW1, const float* __restrict__ gb1,
    const float* __restrict__ gW2, const float* __restrict__ gb2,
    const float* __restrict__ fW1, const float* __restrict__ fb1,
    const float* __restrict__ fW2, const float* __restrict__ fb2,
    float* __restrict__ out) {
    __shared__ __align__(16) float WB[4608]; // gW1 chunk (64x32) -> later gW2|fW1
    __shared__ float BB[128];                // gb1 | gb2 | fb1 | fb2
    __shared__ __align__(16) float P[4][2304];

    const int tid = threadIdx.x, w = tid >> 5, lane = tid & 31;
    const int l15 = lane & 15, hi = lane >> 4;
    const int base = (blockIdx.x * 4 + w) * 16;

    if (tid < 32) { BB[tid] = gb1[tid]; BB[32 + tid] = gb2[tid]; BB[64 + tid] = fb1[tid]; }
    if (tid == 0) BB[96] = fb2[0];

    v8f z8  = {0.f, 0.f, 0.f, 0.f, 0.f, 0.f, 0.f, 0.f};
    v8f ae0 = z8, ae1 = z8, aa0 = z8, aa1 = z8;

    // ego/agent (16x1280) @ gW1 (1280x32), streamed over T chunks of K=64
    for (int t = 0; t < T_STEPS; ++t) {
        __syncthreads();
        for (int idx = tid; idx < 2048; idx += 128)
            WB[idx] = gW1[(size_t)(t * 64 + (idx >> 5)) * 32 + (idx & 31)];
        for (int idx = lane; idx < 1024; idx += 32) {
            int bb = idx >> 6, k = idx & 63;
            size_t re = ((size_t)t * BN + (size_t)(base + bb) * 16) * 32;     // node 0
            size_t ra = ((size_t)t * BN + (size_t)(base + bb) * 16 + 1) * 32; // node 1
            P[w][idx]        = (k < 32) ? hs1[re + k] : hs2[re + k - 32];
            P[w][1024 + idx] = (k < 32) ? hs1[ra + k] : hs2[ra + k - 32];
        }
        lds_fence();
        __syncthreads();
        ae0 = wmma_acc16(ae0, &P[w][0],    64, WB,      32, 64);
        ae1 = wmma_acc16(ae1, &P[w][0],    64, WB + 16, 32, 64);
        aa0 = wmma_acc16(aa0, &P[w][1024], 64, WB,      32, 64);
        aa1 = wmma_acc16(aa1, &P[w][1024], 64, WB + 16, 32, 64);
    }
    __syncthreads();
    // ue/ua = relu(acc + gb1): P[0..511] / P[512..1023]
#pragma unroll
    for (int r = 0; r < 8; ++r) {
        int row = r + 8 * hi;
        float c0v = BB[l15], c1v = BB[16 + l15];
        P[w][row * 32 + l15]            = fmaxf(ae0[r] + c0v, 0.f);
        P[w][row * 32 + 16 + l15]       = fmaxf(ae1[r] + c1v, 0.f);
        P[w][512 + row * 32 + l15]      = fmaxf(aa0[r] + c0v, 0.f);
        P[w][512 + row * 32 + 16 + l15] = fmaxf(aa1[r] + c1v, 0.f);
    }
    lds_fence();
    __syncthreads();
    for (int idx = tid; idx < 4608; idx += 128)
        WB[idx] = (idx < 1024) ? gW2[idx] : fW1[idx - 1024];
    __syncthreads();
    // c_feat = relu(ue@gW2 + gb2) - relu(ua@gW2 + gb2)  -> P[1792..2303]
#pragma unroll
    for (int nt = 0; nt < 2; ++nt) {
        v8f de = wmma_acc16(z8, &P[w][0],   32, WB + 16 * nt, 32, 32);
        v8f da = wmma_acc16(z8, &P[w][512], 32, WB + 16 * nt, 32, 32);
        int c = 16 * nt + l15;
        float bv = BB[32 + c];
#pragma unroll
        for (int r = 0; r < 8; ++r)
            P[w][1792 + (r + 8 * hi) * 32 + c] =
                fmaxf(de[r] + bv, 0.f) - fmaxf(da[r] + bv, 0.f);
    }
    lds_fence();
    // in_f = [x_pair(80) | c_feat(32)]  -> P[0..1791]
    for (int idx = lane; idx < 1792; idx += 32) {
        int bb = idx / 112, c = idx % 112;
        P[w][idx] = (c < 80) ? x_pair[(size_t)(base + bb) * 80 + c]
                             : P[w][1792 + bb * 32 + (c - 80)];
    }
    lds_fence();
    // f1 = relu(in_f @ fW1 + fb1) -> P[1792..2303]
#pragma unroll
    for (int nt = 0; nt < 2; ++nt) {
        v8f f = wmma_acc16(z8, &P[w][0], 112, WB + 1024 + 16 * nt, 32, 112);
        int c = 16 * nt + l15;
        float bv = BB[64 + c];
#pragma unroll
        for (int r = 0; r < 8; ++r)
            P[w][1792 + (r + 8 * hi) * 32 + c] = fmaxf(f[r] + bv, 0.f);
    }
    lds_fence();
    if (lane < 16) {
        float s = BB[96];
        for (int j = 0; j < 32; ++j) s += P[w][1792 + lane * 32 + j] * fW2[j];
        out[(size_t)(base + lane)] = 1.f / (1.f + __expf(-s));
    }
}

// ================================ launch ================================
extern "C" void kernel_launch(void* const* d_in, const int* in_sizes, int n_in,
                              void* d_out, int out_size, void* d_ws, size_t ws_size,
                              hipStream_t stream) {
    const float* x_pair = (const float*)d_in[0];
    const float* x      = (const float*)d_in[1];
    const float* h0     = (const float*)d_in[2];
    const float* c0     = (const float*)d_in[3];
    const float* h0g    = (const float*)d_in[4];
    const float* c0g    = (const float*)d_in[5];
    const float* lWih   = (const float*)d_in[6];
    const float* lWhh   = (const float*)d_in[7];
    const float* lbih   = (const float*)d_in[8];
    const float* lbhh   = (const float*)d_in[9];
    const float* gWih   = (const float*)d_in[10];
    const float* gWhh   = (const float*)d_in[11];
    const float* gbih   = (const float*)d_in[12];
    const float* gbhh   = (const float*)d_in[13];
    const float* w0     = (const float*)d_in[14];
    const float* as0    = (const float*)d_in[15];
    const float* ad0    = (const float*)d_in[16];
    const float* b0     = (const float*)d_in[17];
    const float* w1     = (const float*)d_in[18];
    const float* as1    = (const float*)d_in[19];
    const float* ad1    = (const float*)d_in[20];
    const float* b1     = (const float*)d_in[21];
    const float* gW1    = (const float*)d_in[22];
    const float* gb1    = (const float*)d_in[23];
    const float* gW2    = (const float*)d_in[24];
    const float* gb2    = (const float*)d_in[25];
    const float* fW1    = (const float*)d_in[26];
    const float* fb1    = (const float*)d_in[27];
    const float* fW2    = (const float*)d_in[28];
    const float* fb2    = (const float*)d_in[29];
    float* out = (float*)d_out;

    const size_t SZ = (size_t)T_STEPS * BN * HID; // 20.97M floats each
    float* hs1   = (float*)d_ws;
    float* graph = hs1 + SZ;
    float* hs2   = graph + SZ;

    // 1) temporal LSTM over (x,y): hs1 (T, BN, 32)
    lstm_kernel<2><<<BN / 32, 64, 0, stream>>>(x, T_STEPS * 2, h0, c0,
                                               lWih, lWhh, lbih, lbhh, hs1);
    // 2) GAT encoder: graph (BN, T, 32)
    gat_kernel<<<MG / 4, 128, 0, stream>>>(hs1, w0, as0, ad0, b0,
                                           w1, as1, ad1, b1, graph);
    // 3) graph LSTM: hs2 (T, BN, 32)
    lstm_kernel<32><<<BN / 32, 64, 0, stream>>>(graph, T_STEPS * HID, h0g, c0g,
                                                gWih, gWhh, gbih, gbhh, hs2);
    // 4) gmlp difference + classifier head
    head_kernel<<<BATCH / 64, 128, 0, stream>>>(hs1, hs2, x_pair,
                                                gW1, gb1, gW2, gb2,
                                                fW1, fb1, fW2, fb2, out);
}